// DetectionLoss_87729001988806
// MI455X (gfx1250) — compile-verified
//
#include <hip/hip_runtime.h>

// Problem constants (match reference)
constexpr int   B_ = 64, Q_ = 1800, T_ = 300, C_ = 256;
constexpr float ALPHA = 1.0f, BETA = 1.0f, IOU_THRESH = 0.1f;

#define BLK 512
#define NWAVE (BLK / 32)

typedef __attribute__((ext_vector_type(2))) float v2f;
typedef __attribute__((ext_vector_type(8))) float v8f;

__global__ __launch_bounds__(BLK, 1)
void det_loss_batch(const float* __restrict__ logits,      // [B,Q,C]
                    const float* __restrict__ pred_boxes,  // [B,Q,4]
                    const float* __restrict__ tgt_boxes,   // [B,T,4]
                    const int*  __restrict__ tgt_labels,   // [B,T]
                    float* __restrict__ partials)          // [B,2] (cls,reg)
{
    // ---- LDS working set (~58 KB) ----
    __shared__ float pbx1[Q_], pby1[Q_], pbx2[Q_], pby2[Q_];
    __shared__ float tbx1[T_], tby1[T_], tbx2[T_], tby2[T_], tarea[T_];
    __shared__ float rowmax_s[Q_], lse_s[Q_], logit0_s[Q_];
    __shared__ int   match_q[T_];
    __shared__ unsigned abits[(Q_ + 31) / 32];
    __shared__ float redv[NWAVE], redr[NWAVE];
    __shared__ int   redi[NWAVE];

    const int tid  = threadIdx.x;
    const int lane = tid & 31;
    const int wid  = tid >> 5;
    const int b    = blockIdx.x;

    // ---- Phase 0: stage boxes + clear assignment bitmap ----
    for (int i = tid; i < Q_; i += BLK) {
        const float4 p = ((const float4*)pred_boxes)[b * Q_ + i];
        pbx1[i] = p.x; pby1[i] = p.y; pbx2[i] = p.z; pby2[i] = p.w;
    }
    for (int i = tid; i < T_; i += BLK) {
        const float4 t4 = ((const float4*)tgt_boxes)[b * T_ + i];
        tbx1[i] = t4.x; tby1[i] = t4.y; tbx2[i] = t4.z; tby2[i] = t4.w;
        tarea[i] = (t4.z - t4.x) * (t4.w - t4.y);
    }
    for (int i = tid; i < (Q_ + 31) / 32; i += BLK) abits[i] = 0u;

    // ---- Phase 1: per-query row max (numerical stability) + logit[q][0] ----
    for (int q = tid; q < Q_; q += BLK) {
        const float4* rp = (const float4*)(logits + (size_t)(b * Q_ + q) * C_);
        float4 v0 = rp[0];
        logit0_s[q] = v0.x;
        float m = fmaxf(fmaxf(v0.x, v0.y), fmaxf(v0.z, v0.w));
        for (int k = 1; k < C_ / 4; ++k) {
            const float4 v = rp[k];
            m = fmaxf(m, fmaxf(fmaxf(v.x, v.y), fmaxf(v.z, v.w)));
        }
        rowmax_s[q] = m;
    }
    __syncthreads();

    // ---- Phase 2: log-sum-exp via V_WMMA_F32_16X16X4_F32 against ones ----
    // Each wave owns a 16-query tile. A (16x4 f32) layout: lanes 0-15 carry
    // K=0,1 of row=lane; lanes 16-31 carry K=2,3. D += A x Ones accumulates
    // the 256-class exp-sum for each row into every accumulator column.
    // Pad rows of the final partial tile are CLAMPED to row Q-1 (their result
    // is computed but never stored) so the inner loop is fully branch-free.
    {
        const int nTiles = (Q_ + 15) / 16;   // 113
        const int row = lane & 15;
        const int h   = lane >> 4;
        v2f ones2; ones2[0] = 1.0f; ones2[1] = 1.0f;
        for (int tile = wid; tile < nTiles; tile += NWAVE) {
            const int qb = tile * 16;
            int q = qb + row;
            q = (q < Q_) ? q : (Q_ - 1);               // clamp pad rows
            const float mq = rowmax_s[q];
            // float2 index i covers classes {2i, 2i+1}; lane needs {c+2h, c+2h+1}
            const float2* rp2 =
                (const float2*)(logits + (size_t)(b * Q_ + q) * C_) + h;
            v8f acc = {0.f, 0.f, 0.f, 0.f, 0.f, 0.f, 0.f, 0.f};
#pragma unroll 4
            for (int c = 0; c < C_; c += 4) {
                const float2 x = rp2[c >> 1];          // classes c+2h, c+2h+1
                v2f a;
                a[0] = __expf(x.x - mq);
                a[1] = __expf(x.y - mq);
                acc = __builtin_amdgcn_wmma_f32_16x16x4_f32(
                        false, a, false, ones2, (short)0, acc, false, false);
            }
            // D layout: VGPR r, lanes 0-15 -> M=r; lanes 16-31 -> M=8+r.
            // Columns identical => lanes 0 and 16 publish the 16 sums.
            const int h8 = h * 8;
            if (row == 0) {
#pragma unroll
                for (int r = 0; r < 8; ++r) {
                    const int qf = qb + h8 + r;
                    if (qf < Q_) lse_s[qf] = __logf(acc[r]) + rowmax_s[qf];
                }
            }
        }
    }
    __syncthreads();

    // ---- Phase 3: sequential greedy matching (critical path, LDS-only) ----
    for (int t = 0; t < T_; ++t) {
        const float ta1 = tbx1[t], tb1 = tby1[t], ta2 = tbx2[t], tb2 = tby2[t];
        const float at  = tarea[t];
        float bestv = -1.0f;
        int   bestq = 0x7fffffff;
        for (int q = tid; q < Q_; q += BLK) {
            const bool asn = (abits[q >> 5] >> (q & 31)) & 1u;
            const float px1 = pbx1[q], py1 = pby1[q], px2 = pbx2[q], py2 = pby2[q];
            const float iw = fminf(px2, ta2) - fmaxf(px1, ta1);
            const float ih = fminf(py2, tb2) - fmaxf(py1, tb1);
            const float inter = fmaxf(iw, 0.0f) * fmaxf(ih, 0.0f);
            const float ap = (px2 - px1) * (py2 - py1);
            float iou = inter / (ap + at - inter + 1e-6f);
            iou = asn ? 0.0f : iou;
            if (iou > bestv) { bestv = iou; bestq = q; }   // ascending q => first-max
        }
        // wave32 argmax reduce; tie -> lowest index (matches jnp.argmax)
#pragma unroll
        for (int off = 16; off; off >>= 1) {
            const float ov = __shfl_xor(bestv, off, 32);
            const int   oq = __shfl_xor(bestq, off, 32);
            if (ov > bestv || (ov == bestv && oq < bestq)) { bestv = ov; bestq = oq; }
        }
        if (lane == 0) { redv[wid] = bestv; redi[wid] = bestq; }
        __syncthreads();
        if (wid == 0) {
            float v  = (lane < NWAVE) ? redv[lane] : -1.0f;
            int   qi = (lane < NWAVE) ? redi[lane] : 0x7fffffff;
#pragma unroll
            for (int off = 16; off; off >>= 1) {
                const float ov = __shfl_xor(v, off, 32);
                const int   oq = __shfl_xor(qi, off, 32);
                if (ov > v || (ov == v && oq < qi)) { v = ov; qi = oq; }
            }
            if (lane == 0) {
                int m = -1;
                if (v > IOU_THRESH) { abits[qi >> 5] |= (1u << (qi & 31)); m = qi; }
                match_q[t] = m;   // CE/L1 work deferred off the critical path
            }
        }
        __syncthreads();
    }

    // ---- Phase 4: parallel loss accumulation (deterministic tree reduce) ----
    float cls = 0.0f, reg = 0.0f;
    for (int t = tid; t < T_; t += BLK) {
        const int q = match_q[t];
        if (q >= 0) {
            const int lab = tgt_labels[b * T_ + t];
            cls += lse_s[q] - logits[(size_t)(b * Q_ + q) * C_ + lab];
            float s = 0.0f, d;
            d = fabsf(pbx1[q] - tbx1[t]); s += (d < 1.0f) ? 0.5f * d * d : d - 0.5f;
            d = fabsf(pby1[q] - tby1[t]); s += (d < 1.0f) ? 0.5f * d * d : d - 0.5f;
            d = fabsf(pbx2[q] - tbx2[t]); s += (d < 1.0f) ? 0.5f * d * d : d - 0.5f;
            d = fabsf(pby2[q] - tby2[t]); s += (d < 1.0f) ? 0.5f * d * d : d - 0.5f;
            reg += 0.25f * s;
        }
    }
    for (int q = tid; q < Q_; q += BLK) {
        if (!((abits[q >> 5] >> (q & 31)) & 1u))
            cls += lse_s[q] - logit0_s[q];
    }
#pragma unroll
    for (int off = 16; off; off >>= 1) {
        cls += __shfl_xor(cls, off, 32);
        reg += __shfl_xor(reg, off, 32);
    }
    if (lane == 0) { redv[wid] = cls; redr[wid] = reg; }
    __syncthreads();
    if (tid == 0) {
        float c = 0.0f, r = 0.0f;
        for (int w = 0; w < NWAVE; ++w) { c += redv[w]; r += redr[w]; }
        partials[2 * b] = c;
        partials[2 * b + 1] = r;
    }
}

__global__ void det_loss_finalize(const float* __restrict__ partials,
                                  float* __restrict__ out)
{
    const int lane = threadIdx.x;
    float c = 0.0f, r = 0.0f;
    for (int b = lane; b < B_; b += 32) {
        c += partials[2 * b];
        r += partials[2 * b + 1];
    }
#pragma unroll
    for (int off = 16; off; off >>= 1) {
        c += __shfl_xor(c, off, 32);
        r += __shfl_xor(r, off, 32);
    }
    if (lane == 0) out[0] = (ALPHA * c + BETA * r) / (float)B_;
}

extern "C" void kernel_launch(void* const* d_in, const int* in_sizes, int n_in,
                              void* d_out, int out_size, void* d_ws, size_t ws_size,
                              hipStream_t stream)
{
    const float* logits = (const float*)d_in[0];   // [B,Q,C] f32
    const float* pboxes = (const float*)d_in[1];   // [B,Q,4] f32
    const float* tboxes = (const float*)d_in[2];   // [B,T,4] f32
    const int*   labels = (const int*)d_in[3];     // [B,T]   i32
    float* partials = (float*)d_ws;                // 64*2 floats scratch

    det_loss_batch<<<B_, BLK, 0, stream>>>(logits, pboxes, tboxes, labels, partials);
    det_loss_finalize<<<1, 32, 0, stream>>>(partials, (float*)d_out);
}